// HungarianMatcher_31267361915399
// MI455X (gfx1250) — compile-verified
//
#include <hip/hip_runtime.h>
#include <hip/hip_bf16.h>
#include <math.h>

// Problem constants (from reference)
#define BS 32
#define NQ 300
#define NC 2001
#define NT 25
#define NROWS (BS * NQ)       // 9600 softmax rows
#define KSPLIT 4
#define KPART 504             // ceil-ish split: 504,504,504,489

typedef __attribute__((ext_vector_type(2))) float v2f;
typedef __attribute__((ext_vector_type(8))) float v8f;

// ---------------------------------------------------------------------------
// Kernel A: partial softmax denominators via V_WMMA_F32_16X16X4_F32.
// One wave per (16-row group, K-part): 600 * 4 = 2400 waves.
// A-tile = exp(logits) 16x4 f32 (ISA layout: lanes 0-15 hold K={0,1},
// lanes 16-31 hold K={2,3}), B = ones -> D columns hold the row sums.
// Branch-free unroll-4 main loop: 8 independent b32 loads in flight per wait
// (row stride 2001 floats is odd, so 4B is the only aligned load width).
// ---------------------------------------------------------------------------
__global__ __launch_bounds__(32)
void hm_denom_wmma_kernel(const float* __restrict__ logits,
                          float* __restrict__ partial) {
  const int lane = threadIdx.x;            // wave32
  const int unit = blockIdx.x;
  const int p    = unit & (KSPLIT - 1);    // K-part
  const int rg   = unit >> 2;              // 16-row group
  const int r0   = rg * 16;
  const int m    = lane & 15;              // A-matrix row for this lane
  const int koff = (lane >> 4) * 2;        // lanes 16-31 hold K=2,3
  const size_t base = (size_t)(r0 + m) * NC + koff;

  const int kbeg      = p * KPART;
  const int kend_full = (p < KSPLIT - 1) ? kbeg + KPART : 2000; // 488 full cols in last part

  v8f acc = {};
  v2f bones; bones.x = 1.0f; bones.y = 1.0f;

  int k0 = kbeg;
  // Main loop: 4 WMMA steps per iteration, all loads issued up front.
  for (; k0 + 16 <= kend_full; k0 += 16) {
    __builtin_prefetch(&logits[base + k0 + 512], 0, 1);  // global_prefetch_b8
    float x0 = logits[base + k0 + 0];
    float x1 = logits[base + k0 + 1];
    float x2 = logits[base + k0 + 4];
    float x3 = logits[base + k0 + 5];
    float x4 = logits[base + k0 + 8];
    float x5 = logits[base + k0 + 9];
    float x6 = logits[base + k0 + 12];
    float x7 = logits[base + k0 + 13];
    v2f a0; a0.x = expf(x0); a0.y = expf(x1);
    v2f a1; a1.x = expf(x2); a1.y = expf(x3);
    v2f a2; a2.x = expf(x4); a2.y = expf(x5);
    v2f a3; a3.x = expf(x6); a3.y = expf(x7);
    acc = __builtin_amdgcn_wmma_f32_16x16x4_f32(false, a0, false, bones, (short)0, acc, false, false);
    acc = __builtin_amdgcn_wmma_f32_16x16x4_f32(false, a1, false, bones, (short)0, acc, false, false);
    acc = __builtin_amdgcn_wmma_f32_16x16x4_f32(false, a2, false, bones, (short)0, acc, false, false);
    acc = __builtin_amdgcn_wmma_f32_16x16x4_f32(false, a3, false, bones, (short)0, acc, false, false);
  }
  // Remainder full steps (504 % 16 == 8, 488 % 16 == 8 -> 2 steps)
  for (; k0 < kend_full; k0 += 4) {
    float x0 = logits[base + k0 + 0];
    float x1 = logits[base + k0 + 1];
    v2f a; a.x = expf(x0); a.y = expf(x1);
    acc = __builtin_amdgcn_wmma_f32_16x16x4_f32(false, a, false, bones, (short)0, acc, false, false);
  }
  // Ragged tail: only column k=2000 exists (last part only; uniform branch).
  if (p == KSPLIT - 1) {
    float e0 = 0.0f;
    if (koff == 0) e0 = expf(logits[base + 2000]);
    v2f a; a.x = e0; a.y = 0.0f;
    acc = __builtin_amdgcn_wmma_f32_16x16x4_f32(false, a, false, bones, (short)0, acc, false, false);
  }

  // C/D layout: acc[r] holds row (r + 8*(lane>=16)); every column is the sum.
  // Lanes 0 and 16 (column N=0 holders) write the 16 partial row sums.
  if ((lane & 15) == 0) {
    const int half = (lane >> 4) * 8;
#pragma unroll
    for (int r = 0; r < 8; ++r) {
      partial[p * NROWS + r0 + half + r] = acc[r];
    }
  }
}

// ---------------------------------------------------------------------------
// Kernel B: cost[b][t][q] = -exp(logit[b,q,targets[b,t]]) / denom[b,q]
// denom = fixed-order sum of the 4 K-part partials (bit-deterministic).
// ---------------------------------------------------------------------------
__global__ __launch_bounds__(256)
void hm_cost_kernel(const float* __restrict__ logits,
                    const int* __restrict__ targets,
                    const float* __restrict__ partial,
                    float* __restrict__ cost) {
  const int idx = blockIdx.x * blockDim.x + threadIdx.x;
  if (idx >= BS * NT * NQ) return;
  const int b   = idx / (NT * NQ);
  const int rem = idx - b * (NT * NQ);
  const int t   = rem / NQ;
  const int q   = rem - t * NQ;
  const int row = b * NQ + q;
  const float denom = ((partial[0 * NROWS + row] + partial[1 * NROWS + row])
                     + partial[2 * NROWS + row]) + partial[3 * NROWS + row];
  const int cls = targets[b * NT + t];
  const float lg = logits[(size_t)row * NC + cls];
  cost[idx] = -expf(lg) / denom;
}

// ---------------------------------------------------------------------------
// Kernel C: Jonker-Volgenant LSAP, one workgroup per batch element.
// Cost tile (25x300 f32 = 30 KB) + all state in LDS. Column phase is parallel
// over 300 columns; bookkeeping/augmentation serial in thread 0. f64 duals
// (reference casts cost to float64). Argmin tie-breaks to lowest column index
// (numpy first-occurrence semantics).
// ---------------------------------------------------------------------------
__global__ __launch_bounds__(512)
void hm_lsap_kernel(const float* __restrict__ cost,
                    int* __restrict__ out) {
  __shared__ float  Cs[NT * NQ];
  __shared__ double sh_short[NQ];
  __shared__ double sh_v[NQ];
  __shared__ double sh_u[NT];
  __shared__ int    sh_path[NQ];
  __shared__ int    sh_r4c[NQ];     // row4col
  __shared__ int    sh_SC[NQ];
  __shared__ int    sh_c4r[NT];     // col4row
  __shared__ double red_val[512];
  __shared__ int    red_idx[512];
  __shared__ int    sh_srR[NT + 1]; // visited rows this augmentation
  __shared__ int    sh_i, sh_sink, sh_nsr;
  __shared__ double sh_minVal;

  const int tid = threadIdx.x;
  const int b   = blockIdx.x;
  const double INF = 1e300;

  for (int idx = tid; idx < NT * NQ; idx += 512)
    Cs[idx] = cost[(size_t)b * NT * NQ + idx];
  if (tid < NQ) { sh_v[tid] = 0.0; sh_r4c[tid] = -1; }
  if (tid < NT) { sh_u[tid] = 0.0; sh_c4r[tid] = -1; }
  __syncthreads();

  for (int cur = 0; cur < NT; ++cur) {
    if (tid < NQ) { sh_short[tid] = INF; sh_path[tid] = -1; sh_SC[tid] = 0; }
    if (tid == 0) { sh_i = cur; sh_minVal = 0.0; sh_sink = -1; sh_nsr = 0; }
    __syncthreads();

    // --- shortest augmenting path (Dijkstra over columns) ---
    while (true) {
      const int    i      = sh_i;
      const double minVal = sh_minVal;

      double val = INF; int idx = -1;
      if (tid < NQ && !sh_SC[tid]) {
        const double d = minVal + (double)Cs[i * NQ + tid] - sh_u[i] - sh_v[tid];
        if (d < sh_short[tid]) { sh_short[tid] = d; sh_path[tid] = i; }
        val = sh_short[tid]; idx = tid;
      }
      red_val[tid] = val; red_idx[tid] = idx;
      __syncthreads();

      for (int s = 256; s > 0; s >>= 1) {
        if (tid < s) {
          const double ov = red_val[tid + s]; const int oi = red_idx[tid + s];
          if (oi != -1 &&
              (red_idx[tid] == -1 || ov < red_val[tid] ||
               (ov == red_val[tid] && oi < red_idx[tid]))) {
            red_val[tid] = ov; red_idx[tid] = oi;
          }
        }
        __syncthreads();
      }

      if (tid == 0) {
        const int jpos = red_idx[0];
        sh_minVal = red_val[0];
        sh_SC[jpos] = 1;
        sh_srR[sh_nsr++] = i;
        if (sh_r4c[jpos] == -1) sh_sink = jpos;
        else                    sh_i = sh_r4c[jpos];
      }
      __syncthreads();
      if (sh_sink != -1) break;
    }

    const double mv   = sh_minVal;
    const int    sink = sh_sink;
    const int    nsr  = sh_nsr;

    // --- dual updates ---
    if (tid == 0) {
      sh_u[cur] += mv;
    } else if (tid < nsr) {
      const int irow = sh_srR[tid];          // visited rows distinct; [0]==cur
      sh_u[irow] += mv - sh_short[sh_c4r[irow]];
    }
    if (tid < NQ && sh_SC[tid]) {
      sh_v[tid] -= mv - sh_short[tid];
    }
    __syncthreads();

    // --- augment along found path (serial) ---
    if (tid == 0) {
      int j = sink;
      while (true) {
        const int ir = sh_path[j];
        sh_r4c[j] = ir;
        const int tmp = sh_c4r[ir];
        sh_c4r[ir] = j;
        j = tmp;
        if (ir == cur) break;
      }
    }
    __syncthreads();
  }

  // --- stable ordering of the 25 assignments by query index (distinct) ---
  // Output (reference dtype int32): row_inds [BS][NT] then col_inds [BS][NT]
  if (tid < NT) {
    const int my = sh_c4r[tid];
    int rank = 0;
#pragma unroll
    for (int t2 = 0; t2 < NT; ++t2) {
      const int o = sh_c4r[t2];
      if (o < my || (o == my && t2 < tid)) ++rank;
    }
    out[b * NT + rank]           = my;   // row_inds: sorted query indices
    out[BS * NT + b * NT + rank] = tid;  // col_inds: target index per slot
  }
}

// ---------------------------------------------------------------------------
extern "C" void kernel_launch(void* const* d_in, const int* in_sizes, int n_in,
                              void* d_out, int out_size, void* d_ws, size_t ws_size,
                              hipStream_t stream) {
  const float* logits  = (const float*)d_in[0];  // [32, 300, 2001] f32
  const int*   targets = (const int*)d_in[1];    // [32, 25] i32

  // Workspace: partial denoms [KSPLIT][9600] f32, cost [32*25*300] f32 (~1.1 MB)
  float* partial = (float*)d_ws;
  float* cost    = partial + KSPLIT * NROWS;

  // A: WMMA exp row-sum partials; 600 row-groups x 4 K-parts = 2400 waves
  hm_denom_wmma_kernel<<<dim3((NROWS / 16) * KSPLIT), dim3(32), 0, stream>>>(
      logits, partial);

  // B: cost gather (+ fixed-order partial reduction)
  const int ncost = BS * NT * NQ;
  hm_cost_kernel<<<dim3((ncost + 255) / 256), dim3(256), 0, stream>>>(
      logits, targets, partial, cost);

  // C: per-batch Hungarian in LDS + output ordering
  hm_lsap_kernel<<<dim3(BS), dim3(512), 0, stream>>>(cost, (int*)d_out);
}